// LMInfinite_74345883893972
// MI455X (gfx1250) — compile-verified
//
#include <hip/hip_runtime.h>

typedef __attribute__((ext_vector_type(2))) float v2f;
typedef __attribute__((ext_vector_type(8))) float v8f;

// Problem constants from the reference
constexpr int S_DIM = 512;   // sequence length (B == Sq == Sk)
constexpr int D_DIM = 128;   // head dim
constexpr int N_LOCAL = 128; // sliding window
constexpr int L_PRETRAIN = 256;

// 1/sqrt(128); reference divides by sqrt(D) in f32.
#define RSCALE 0.08838834764831845f

// Compute a 16 x (NT*16) logit strip with V_WMMA_F32_16X16X4_F32.
// qp: &Q[b, r0+l15, half*2]  (A fragment: lane l15 = row, half selects K pair)
// kb: &K[b, cbase+l15, half*2] (B fragment: lane l15 = column n of tile 0)
template <int NT>
__device__ __forceinline__ void compute_strip(const float* __restrict__ qp,
                                              const float* __restrict__ kb,
                                              v8f acc[NT]) {
#pragma unroll 4
  for (int ks = 0; ks < D_DIM / 4; ++ks) {
    const v2f a = *(const v2f*)(qp + (ks << 2));
#pragma unroll
    for (int t = 0; t < NT; ++t) {
      const v2f kv = *(const v2f*)(kb + (size_t)(t * 16) * D_DIM + (ks << 2));
      v2f bv;
      bv.x = floorf(kv.x * RSCALE);
      bv.y = floorf(kv.y * RSCALE);
      acc[t] = __builtin_amdgcn_wmma_f32_16x16x4_f32(
          /*neg_a=*/false, a, /*neg_b=*/false, bv,
          /*c_mod=*/(short)0, acc[t], /*reuse_a=*/false, /*reuse_b=*/false);
    }
  }
}

// Epilogue: out = logit * (mask / divider). Identical to (logit/divider)*mask,
// including row-511 divider==0 NaN/Inf semantics inside computed tiles.
template <int NT>
__device__ __forceinline__ void store_strip(float* __restrict__ ob, int r0,
                                            int cbase, int l15, int half,
                                            const v8f acc[NT]) {
#pragma unroll
  for (int t = 0; t < NT; ++t) {
    const int cn = cbase + t * 16 + l15;
#pragma unroll
    for (int r = 0; r < 8; ++r) {
      const int m = r0 + r + (half << 3);  // C layout: vgpr r -> rows r / r+8
      const int dv = (S_DIM - 1 - m < L_PRETRAIN) ? (S_DIM - 1 - m) : L_PRETRAIN;
      const float divider = (float)dv;
      const float mask =
          ((m < N_LOCAL) || (cn >= m - N_LOCAL && cn < m)) ? 1.0f : 0.0f;
      ob[(size_t)m * S_DIM + cn] = acc[t][r] * (mask / divider);
    }
  }
}

__global__ __launch_bounds__(256) void
LMInfinite_74345883893972_kernel(const float* __restrict__ Q,
                                 const float* __restrict__ K,
                                 float* __restrict__ O) {
  const int b    = blockIdx.x >> 5;  // batch index (SGPR -> uniform branches)
  const int it   = blockIdx.x & 31;  // 16-row tile index
  const int r0   = it << 4;
  const int wave = threadIdx.x >> 5; // 0..7
  const int lane = threadIdx.x & 31;
  const int half = lane >> 4;        // 0: lanes 0-15, 1: lanes 16-31
  const int l15  = lane & 15;

  const float* qp = Q + ((size_t)b * S_DIM + (r0 + l15)) * D_DIM + (half << 1);
  const float* kB = K + (size_t)b * S_DIM * D_DIM + (half << 1);
  float* ob = O + (size_t)b * S_DIM * S_DIM;

  if (r0 < N_LOCAL) {
    // Dense rows: all 512 columns live. 8 waves x 4 tiles.
    const int cbase = wave << 6;
    v8f acc[4] = {};
    compute_strip<4>(qp, kB + (size_t)(cbase + l15) * D_DIM, acc);
    store_strip<4>(ob, r0, cbase, l15, half, acc);
  } else {
    // Banded rows: live band = cols [r0-128, r0+16) (9 tiles). Compute a
    // 256-col window around it (8 waves x 2 tiles); epilogue mask zeroes the
    // in-window dead columns. cstart is clamped so the window stays in-range.
    const int cstart = (r0 - N_LOCAL < S_DIM - 256) ? (r0 - N_LOCAL) : (S_DIM - 256);
    const int cbase  = cstart + (wave << 5);
    v8f acc[2] = {};
    compute_strip<2>(qp, kB + (size_t)(cbase + l15) * D_DIM, acc);
    store_strip<2>(ob, r0, cbase, l15, half, acc);

    // Zero-fill the 256 columns outside the window: [0,cstart) u [cstart+256,512).
    // cstart is a multiple of 16, so float4 chunks never straddle the seam.
    const int tid = threadIdx.x;
#pragma unroll
    for (int i = 0; i < 4; ++i) {
      const int ch  = tid + (i << 8);     // 0..1023: 16 rows x 64 float4 chunks
      const int row = ch >> 6;
      const int c4  = (ch & 63) << 2;     // 0..252, zero-region flat col
      const int col = c4 + ((c4 < cstart) ? 0 : 256);
      float4 z = make_float4(0.f, 0.f, 0.f, 0.f);
      *(float4*)(ob + (size_t)(r0 + row) * S_DIM + col) = z;
    }
  }
}

extern "C" void kernel_launch(void* const* d_in, const int* in_sizes, int n_in,
                              void* d_out, int out_size, void* d_ws, size_t ws_size,
                              hipStream_t stream) {
  const float* q = (const float*)d_in[0];
  const float* k = (const float*)d_in[1];
  float* out = (float*)d_out;

  const int B = in_sizes[0] / (S_DIM * D_DIM);  // 512
  const int rowTiles = S_DIM / 16;              // 32
  dim3 grid((unsigned)(B * rowTiles));
  dim3 block(256);
  LMInfinite_74345883893972_kernel<<<grid, block, 0, stream>>>(q, k, out);
}